// HGNAM_48816598286355
// MI455X (gfx1250) — compile-verified
//
#include <hip/hip_runtime.h>

// Problem dims (fixed by the reference)
#define N_NODES 2048
#define F_FEAT  128
#define H_DIM   64
#define O_DIM   16
#define KSLICES 4                    // K-split of the m-dimension for occupancy

typedef float v2f __attribute__((ext_vector_type(2)));
typedef float v8f __attribute__((ext_vector_type(8)));

// ---------------------------------------------------------------------------
// Kernel 1: f_sums[n,o] = sum_f ( relu(x[n,f]*fW1[f,:]+fb1[f,:]) @ fW2[f,:,:] )
//                         + sum_f fb2[f,o]
// One block = one 16-row n-tile. 8 waves split the 128 features (16 each).
// H=64 contraction = 16 chained V_WMMA_F32_16X16X4_F32 per feature.
// A-layout (f32 16x4): lanes 0-15 -> M=lane, K={0,1}; lanes 16-31 -> M=lane-16, K={2,3}.
// B-layout (f32 4x16): lanes 0-15 -> K={0,1} (vgpr0,1), N=lane; lanes 16-31 -> K={2,3}, N=lane-16.
// Loads are batched per 8-chunk group so the WMMA chain doesn't stall on
// just-in-time B loads (removes per-WMMA s_wait_loadcnt 0).
// ---------------------------------------------------------------------------
__global__ __launch_bounds__(256) void hgnam_fsums_kernel(
    const float* __restrict__ x,    // [N,F]
    const float* __restrict__ fW1,  // [F,H]
    const float* __restrict__ fb1,  // [F,H]
    const float* __restrict__ fW2,  // [F,H,O]
    const float* __restrict__ fb2,  // [F,O]
    float* __restrict__ fsums)      // [N,O]
{
    __shared__ float red[8][256];

    const int lane = threadIdx.x & 31;
    const int wv   = threadIdx.x >> 5;   // 0..7
    const int half = lane >> 4;          // 0 or 1
    const int r    = lane & 15;          // row-in-tile (A) / N column (B)
    const int base_n = blockIdx.x * 16;

    v8f acc = {};

    const int f0 = wv * 16;
    for (int fi = 0; fi < 16; ++fi) {
        const int f = f0 + fi;
        const float xv = x[(base_n + r) * F_FEAT + f];
        const float* __restrict__ w1 = fW1 + f * H_DIM;
        const float* __restrict__ b1 = fb1 + f * H_DIM;
        const float* __restrict__ w2 = fW2 + f * H_DIM * O_DIM;

        #pragma unroll
        for (int g = 0; g < 2; ++g) {               // 2 groups of 8 K-chunks
            v2f av[8], bv[8];
            // ---- load phase: batch all loads for this group ----
            #pragma unroll
            for (int c = 0; c < 8; ++c) {
                const int h0 = 4 * (g * 8 + c) + 2 * half;
                av[c] = *(const v2f*)(w1 + h0);     // fW1 pair (8B aligned)
                bv[c].x = w2[(h0    ) * O_DIM + r];
                bv[c].y = w2[(h0 + 1) * O_DIM + r];
            }
            v2f b1v[8];
            #pragma unroll
            for (int c = 0; c < 8; ++c) {
                const int h0 = 4 * (g * 8 + c) + 2 * half;
                b1v[c] = *(const v2f*)(b1 + h0);
            }
            // ---- compute phase: activations then WMMA chain ----
            #pragma unroll
            for (int c = 0; c < 8; ++c) {
                v2f a;
                a.x = fmaxf(fmaf(xv, av[c].x, b1v[c].x), 0.0f);
                a.y = fmaxf(fmaf(xv, av[c].y, b1v[c].y), 0.0f);
                acc = __builtin_amdgcn_wmma_f32_16x16x4_f32(
                    false, a, false, bv[c], (short)0, acc, false, false);
            }
        }
    }

    // Deterministic 8-wave reduction through LDS, then add fb2 bias and store.
    #pragma unroll
    for (int v = 0; v < 8; ++v) red[wv][lane * 8 + v] = acc[v];
    __syncthreads();
    if (wv == 0) {
        float bias = 0.0f;
        for (int f = 0; f < F_FEAT; ++f) bias += fb2[f * O_DIM + r];
        #pragma unroll
        for (int v = 0; v < 8; ++v) {
            float s = 0.0f;
            #pragma unroll
            for (int u = 0; u < 8; ++u) s += red[u][lane * 8 + v];
            // C/D layout: vgpr v, half -> row M = v + 8*half, col N = r
            fsums[(base_n + v + 8 * half) * O_DIM + r] = s + bias;
        }
    }
}

// ---------------------------------------------------------------------------
// Kernel 2: partial[ks] = (m(dist)/norm)[:, ks-slice] @ f_sums[ks-slice, :]
// grid = (128 n-tiles, KSLICES k-slices); block = 8 waves; each wave owns a
// (2048/KSLICES/8)-wide m-chunk. Per 16x16 k-tile a lane computes exactly the
// 8 m_norm elements it owns in the f32 WMMA A layout (pairs of adjacent
// columns -> aligned float2 loads), evaluates the shared 64-h MLP with
// broadcast weights from LDS (packed f32 FMAs), multiplies by v_rcp_f32 of
// the normalizer, and issues 4 chained WMMAs against pre-loaded f_sums
// B tiles. m_norm is never materialized.
// ---------------------------------------------------------------------------
__global__ __launch_bounds__(256) void hgnam_main_kernel(
    const float* __restrict__ dist,   // [N,N]
    const float* __restrict__ nrm,    // [N,N]
    const float* __restrict__ mW1,    // [H]
    const float* __restrict__ mb1,    // [H]
    const float* __restrict__ mW2,    // [H]
    const float* __restrict__ mb2p,   // [1]
    const float* __restrict__ fsums,  // [N,O]
    float* __restrict__ part)         // [KSLICES, N, O]
{
    __shared__ float sW1[H_DIM], sB1[H_DIM], sW2[H_DIM];
    __shared__ float red[8][256];

    const int lane = threadIdx.x & 31;
    const int wv   = threadIdx.x >> 5;
    const int half = lane >> 4;
    const int r    = lane & 15;
    const int base_n = blockIdx.x * 16;
    const int kbase  = blockIdx.y * (N_NODES / KSLICES) + wv * (N_NODES / KSLICES / 8);
    const int NTILES = N_NODES / KSLICES / 8 / 16;   // k-tiles per wave

    if (threadIdx.x < H_DIM) {
        sW1[threadIdx.x] = mW1[threadIdx.x];
        sB1[threadIdx.x] = mb1[threadIdx.x];
        sW2[threadIdx.x] = mW2[threadIdx.x];
    }
    __syncthreads();

    const float mb2 = mb2p[0];
    const int n = base_n + r;                       // A row M == r for both halves
    const float* __restrict__ drow = dist + (size_t)n * N_NODES;
    const float* __restrict__ vrow = nrm  + (size_t)n * N_NODES;

    v8f acc = {};

    for (int kt = 0; kt < NTILES; ++kt) {
        const int m0 = kbase + kt * 16;

        // Batch-load: dist/norm pairs this lane owns + f_sums B operands,
        // so their latency hides behind the 64-h MLP below.
        v2f d[4], q[4], s[4], bb[4];
        #pragma unroll
        for (int kk = 0; kk < 4; ++kk) {
            const int mm = m0 + 4 * kk + 2 * half;  // 8-byte aligned
            d[kk] = *(const v2f*)(drow + mm);
            q[kk] = *(const v2f*)(vrow + mm);
            bb[kk].x = fsums[(mm    ) * O_DIM + r]; // B[k][N=o=r]
            bb[kk].y = fsums[(mm + 1) * O_DIM + r];
            s[kk].x = mb2; s[kk].y = mb2;
        }

        // Shared m-MLP: 64 hidden units, weights broadcast to all 8 elements.
        #pragma unroll 8
        for (int h = 0; h < H_DIM; ++h) {
            const float w1 = sW1[h], b1 = sB1[h], w2 = sW2[h];
            #pragma unroll
            for (int kk = 0; kk < 4; ++kk) {
                float t0 = fmaxf(fmaf(d[kk].x, w1, b1), 0.0f);
                float t1 = fmaxf(fmaf(d[kk].y, w1, b1), 0.0f);
                s[kk].x = fmaf(t0, w2, s[kk].x);
                s[kk].y = fmaf(t1, w2, s[kk].y);
            }
        }

        // Normalize (v_rcp_f32: norm is bounded in [0.5,1.5]) and contract.
        #pragma unroll
        for (int kk = 0; kk < 4; ++kk) {
            v2f a;
            a.x = s[kk].x * __builtin_amdgcn_rcpf(q[kk].x);
            a.y = s[kk].y * __builtin_amdgcn_rcpf(q[kk].y);
            acc = __builtin_amdgcn_wmma_f32_16x16x4_f32(
                false, a, false, bb[kk], (short)0, acc, false, false);
        }
    }

    // Deterministic 8-wave reduction; one partial tile per (n-tile, k-slice).
    #pragma unroll
    for (int v = 0; v < 8; ++v) red[wv][lane * 8 + v] = acc[v];
    __syncthreads();
    if (wv == 0) {
        float* __restrict__ pslice = part + (size_t)blockIdx.y * (N_NODES * O_DIM);
        #pragma unroll
        for (int v = 0; v < 8; ++v) {
            float ssum = 0.0f;
            #pragma unroll
            for (int u = 0; u < 8; ++u) ssum += red[u][lane * 8 + v];
            pslice[(base_n + v + 8 * half) * O_DIM + r] = ssum;
        }
    }
}

// ---------------------------------------------------------------------------
// Kernel 3: deterministic combine of the KSLICES partials into d_out.
// ---------------------------------------------------------------------------
__global__ __launch_bounds__(256) void hgnam_combine_kernel(
    const float* __restrict__ part, float* __restrict__ out)
{
    const int i = blockIdx.x * 256 + threadIdx.x;
    if (i < N_NODES * O_DIM) {
        float s = 0.0f;
        #pragma unroll
        for (int k = 0; k < KSLICES; ++k)
            s += part[(size_t)k * (N_NODES * O_DIM) + i];
        out[i] = s;
    }
}

extern "C" void kernel_launch(void* const* d_in, const int* in_sizes, int n_in,
                              void* d_out, int out_size, void* d_ws, size_t ws_size,
                              hipStream_t stream) {
    (void)in_sizes; (void)n_in; (void)out_size; (void)ws_size;
    const float* x    = (const float*)d_in[0];
    const float* dist = (const float*)d_in[1];
    const float* nrm  = (const float*)d_in[2];
    const float* fW1  = (const float*)d_in[3];
    const float* fb1  = (const float*)d_in[4];
    const float* fW2  = (const float*)d_in[5];
    const float* fb2  = (const float*)d_in[6];
    const float* mW1  = (const float*)d_in[7];
    const float* mb1  = (const float*)d_in[8];
    const float* mW2  = (const float*)d_in[9];
    const float* mb2  = (const float*)d_in[10];

    float* fsums = (float*)d_ws;                         // 128 KB
    float* part  = (float*)d_ws + N_NODES * O_DIM;       // KSLICES * 128 KB
    float* out   = (float*)d_out;

    hgnam_fsums_kernel<<<dim3(N_NODES / 16), dim3(256), 0, stream>>>(
        x, fW1, fb1, fW2, fb2, fsums);
    hgnam_main_kernel<<<dim3(N_NODES / 16, KSLICES), dim3(256), 0, stream>>>(
        dist, nrm, mW1, mb1, mW2, mb2, fsums, part);
    hgnam_combine_kernel<<<dim3((N_NODES * O_DIM + 255) / 256), dim3(256), 0, stream>>>(
        part, out);
}